// QAPPolicy_72078141162184
// MI455X (gfx1250) — compile-verified
//
#include <hip/hip_runtime.h>
#include <hip/hip_bf16.h>

// Problem constants (from reference)
#define BB 128
#define TT 512
#define NN 512
#define KK 8
#define CAPACITY 50.0f
#define NEG_INF -1000000000.0f

typedef __attribute__((ext_vector_type(2))) float v2f;
typedef __attribute__((ext_vector_type(8))) float v8f;

__global__ __launch_bounds__(256) void qap_eval_kernel(
    const float* __restrict__ psi,      // (B,N1,2)
    const float* __restrict__ dem_g,    // (B,N1)
    const float* __restrict__ Wq_w,     // (2,4)
    const float* __restrict__ Wq_b,     // (2,)
    const float* __restrict__ lam_p,    // (1,)
    const int*   __restrict__ actions,  // (B,T)
    const int*   __restrict__ knn,      // (B,N1,K)
    float*       __restrict__ out)      // lp_new (B,T) then entropy (B)
{
    const int b   = blockIdx.x;
    const int tid = threadIdx.x;

    __shared__ float s_psi0[NN], s_psi1[NN], s_dem[NN], s_c[NN];
    __shared__ float s_q0[TT], s_q1[TT], s_rem[TT], s_entrow[TT];
    __shared__ int   s_act[TT], s_fv[NN];

    // ---- Phase A: stage per-batch data in LDS -------------------------------
    for (int n = tid; n < NN; n += 256) {
        s_psi0[n] = psi[((size_t)b * NN + n) * 2 + 0];
        s_psi1[n] = psi[((size_t)b * NN + n) * 2 + 1];
        s_dem[n]  = dem_g[(size_t)b * NN + n];
        s_act[n]  = actions[(size_t)b * TT + n];   // T == N1 == 512
        s_fv[n]   = 0x40000000;                    // "never visited"
    }
    __syncthreads();

    // first-visit time per node: vis[t,n] = (fv[n] < t), n != 0
    for (int t = tid; t < TT; t += 256) atomicMin(&s_fv[s_act[t]], t);

    // interference term c[n] = lambda * sum_k psi[n] . psi[knn[n,k]]
    const float lam = lam_p[0];
    __syncthreads();
    for (int n = tid; n < NN; n += 256) {
        const float p0 = s_psi0[n], p1 = s_psi1[n];
        const int* kn = knn + ((size_t)b * NN + n) * KK;
        float acc = 0.f;
#pragma unroll
        for (int k = 0; k < KK; ++k) {
            const int j = kn[k];
            acc += p0 * s_psi0[j] + p1 * s_psi1[j];
        }
        s_c[n] = lam * acc;
    }

    // used-capacity recurrence (prefix sum with resets): sequential, tiny.
    if (tid == 0) {
        float used = 0.f;
        s_rem[0] = CAPACITY;
        for (int t = 1; t < TT; ++t) {
            const int ap = s_act[t - 1];
            used = (ap == 0) ? 0.f : used + s_dem[ap];
            s_rem[t] = CAPACITY - used;
        }
    }
    __syncthreads();

    // query q[t] = Wq_w @ ctx + Wq_b,  ctx = [psi_cur0, psi_cur1, cap_norm, t_norm]
    {
        const float w00 = Wq_w[0], w01 = Wq_w[1], w02 = Wq_w[2], w03 = Wq_w[3];
        const float w10 = Wq_w[4], w11 = Wq_w[5], w12 = Wq_w[6], w13 = Wq_w[7];
        const float qb0 = Wq_b[0], qb1 = Wq_b[1];
        for (int t = tid; t < TT; t += 256) {
            const int cur = (t == 0) ? 0 : s_act[t - 1];
            const bool adep = (cur == 0);
            const float pc0 = adep ? 0.f : s_psi0[cur];
            const float pc1 = adep ? 0.f : s_psi1[cur];
            const float capn = s_rem[t] / CAPACITY;
            const float tn   = (float)t / 511.0f;   // / max(n_cust,1)
            s_q0[t] = w00 * pc0 + w01 * pc1 + w02 * capn + w03 * tn + qb0;
            s_q1[t] = w10 * pc0 + w11 * pc1 + w12 * capn + w13 * tn + qb1;
        }
    }
    __syncthreads();

    // ---- Phase B: 16x16 score tiles via V_WMMA_F32_16X16X4_F32 --------------
    // score[t,n] = q0[t]*psi0[n] + q1[t]*psi1[n] + 1*c[n] + 0  (K = 4 exactly)
    const int wave = tid >> 5;
    const int lane = tid & 31;
    const int hl   = lane & 15;
    const bool hi  = lane >= 16;

    for (int tt = wave; tt < TT / 16; tt += 8) {
        const int t0 = tt * 16;

        // A (16x4): lanes 0-15 -> K=0,1 ; lanes 16-31 -> K=2,3
        v2f A;
        A[0] = hi ? 1.0f : s_q0[t0 + hl];
        A[1] = hi ? 0.0f : s_q1[t0 + hl];

        // Per-lane row constants (row R = r + (hi ? 8 : 0))
        int   rowt[8];
        float rowrem[8];
#pragma unroll
        for (int r = 0; r < 8; ++r) {
            const int R = r + (hi ? 8 : 0);
            rowt[r]   = t0 + R;
            rowrem[r] = s_rem[t0 + R];
        }

        // Online-softmax accumulators per row: max m, sum e^{x-m}, sum x e^{x-m}
        float am[8], ase[8], asx[8];
        unsigned hc = 0;  // bit r: row has an unmasked customer (n>=1)
#pragma unroll
        for (int r = 0; r < 8; ++r) { am[r] = -3.0e38f; ase[r] = 0.f; asx[r] = 0.f; }

        for (int nt = 0; nt < NN / 16; ++nt) {
            const int nl = nt * 16 + hl;
            // B (4x16): lanes 0-15 -> K=0,1 ; lanes 16-31 -> K=2,3
            v2f Bm;
            Bm[0] = hi ? s_c[nl]   : s_psi0[nl];
            Bm[1] = hi ? 0.0f      : s_psi1[nl];
            v8f C = {};
            v8f D = __builtin_amdgcn_wmma_f32_16x16x4_f32(
                false, A, false, Bm, (short)0, C, false, false);

            const float dn = s_dem[nl];
            const int   fv = s_fv[nl];
#pragma unroll
            for (int r = 0; r < 8; ++r) {
                float x = D[r];
                bool masked;
                if (nl == 0) masked = true;                 // depot merged at finalize
                else         masked = (fv < rowt[r]) || (dn > rowrem[r]);
                if (!masked && nl != 0) hc |= (1u << r);
                x = masked ? NEG_INF : x;
                const float mnew  = fmaxf(am[r], x);
                const float scale = __expf(am[r] - mnew);
                const float e     = __expf(x - mnew);
                ase[r] = ase[r] * scale + e;
                asx[r] = asx[r] * scale + e * x;
                am[r]  = mnew;
            }
        }

        // merge online-softmax states across the 16-lane half (order-independent)
#pragma unroll
        for (int off = 1; off < 16; off <<= 1) {
#pragma unroll
            for (int r = 0; r < 8; ++r) {
                const float om  = __shfl_xor(am[r], off);
                const float ose = __shfl_xor(ase[r], off);
                const float osx = __shfl_xor(asx[r], off);
                const float mnew = fmaxf(am[r], om);
                const float s1 = __expf(am[r] - mnew);
                const float s2 = __expf(om - mnew);
                ase[r] = ase[r] * s1 + ose * s2;
                asx[r] = asx[r] * s1 + osx * s2;
                am[r]  = mnew;
            }
            hc |= (unsigned)__shfl_xor((int)hc, off);
        }

        // finalize: lane hl==r of each half owns row R = r + (hi?8:0)
        if (hl < 8) {
            const int r = hl;
            const int t = rowt[r];
            float m = am[r], se = ase[r], sx = asx[r];
            const bool adep    = (t == 0) || (s_act[t - 1] == 0);
            const bool hascust = (hc >> r) & 1u;
            // depot element n=0: mask = at_depot && has_cust
            const float s0v = s_q0[t] * s_psi0[0] + s_q1[t] * s_psi1[0] + s_c[0];
            const bool  mask0 = adep && hascust;
            const float x0 = mask0 ? NEG_INF : s0v;
            {
                const float mnew = fmaxf(m, x0);
                const float s1 = __expf(m - mnew);
                const float e  = __expf(x0 - mnew);
                se = se * s1 + e;
                sx = sx * s1 + e * x0;
                m  = mnew;
            }
            const float lse = m + __logf(se);
            const float ent = lse - sx / se;          // -sum p*logp
            const int a = s_act[t];
            float xa;
            if (a == 0) {
                xa = x0;
            } else {
                const bool masked = (s_fv[a] < t) || (s_dem[a] > s_rem[t]);
                xa = masked ? NEG_INF
                            : (s_q0[t] * s_psi0[a] + s_q1[t] * s_psi1[a] + s_c[a]);
            }
            out[(size_t)b * TT + t] = xa - lse;
            s_entrow[t] = ent;
        }
    }
    __syncthreads();

    // deterministic entropy mean reduction
    __shared__ float s_red[256];
    float v = s_entrow[tid] + s_entrow[tid + 256];
    s_red[tid] = v;
    __syncthreads();
    for (int s = 128; s > 0; s >>= 1) {
        if (tid < s) s_red[tid] += s_red[tid + s];
        __syncthreads();
    }
    if (tid == 0) out[(size_t)BB * TT + b] = s_red[0] * (1.0f / (float)TT);
}

extern "C" void kernel_launch(void* const* d_in, const int* in_sizes, int n_in,
                              void* d_out, int out_size, void* d_ws, size_t ws_size,
                              hipStream_t stream) {
    const float* psi     = (const float*)d_in[0];
    const float* demands = (const float*)d_in[1];
    const float* Wq_w    = (const float*)d_in[2];
    const float* Wq_b    = (const float*)d_in[3];
    const float* lam     = (const float*)d_in[4];
    const int*   actions = (const int*)d_in[5];
    const int*   knn     = (const int*)d_in[6];
    float* out = (float*)d_out;

    qap_eval_kernel<<<dim3(BB), dim3(256), 0, stream>>>(
        psi, demands, Wq_w, Wq_b, lam, actions, knn, out);
}